// RBFmodel_55765855371808
// MI455X (gfx1250) — compile-verified
//
#include <hip/hip_runtime.h>

typedef __attribute__((ext_vector_type(2))) float v2f;
typedef __attribute__((ext_vector_type(8))) float v8f;

#define ROWS 128
#define COLS 128
#define N_IMAGES 32
#define N_GAUSS 128

// One wave32 handles a 16-pixel tile (one row, 16 consecutive cols) of one image.
// Q(16x16) = F(16x6) * G^T(6x16) via two V_WMMA_F32_16X16X4_F32 steps (K padded to 8),
// then z = sum_N exp2(Q) * amp[N] with a shfl_xor butterfly over the 16-lane N groups.
__global__ __launch_bounds__(256) void rbf_wmma_kernel(
    const float* __restrict__ centers,      // [32][128][2]
    const float* __restrict__ covs,         // [32][128][3]
    const float* __restrict__ amps,         // [32][128][1]
    float* __restrict__ out)                // [32][128][128]
{
    __shared__ float stage[8 * 16];         // 8 waves per block, 16 pixels each

    const unsigned tid       = threadIdx.x;
    const unsigned lane      = tid & 31u;
    const unsigned waveInBlk = tid >> 5;
    const unsigned wid       = blockIdx.x * 8u + waveInBlk;  // 0..32767
    const unsigned img       = wid >> 10;                    // wid / 1024
    const unsigned tile      = wid & 1023u;                  // pixel tile in image
    const unsigned row       = tile >> 3;                    // 0..127
    const unsigned c0        = (tile & 7u) * 16u;            // column base
    const unsigned m         = lane & 15u;
    const bool     hi        = lane >= 16u;

    // ---- A operand: per-pixel features, analytic grid (no memory traffic) ----
    // out[n,r,c] uses gx = lin(c), gy = lin(r)  (from reference reshape+swapaxes)
    const float gx = (float)(c0 + m) * (2.0f / 127.0f) - 1.0f;
    const float gy = (float)row      * (2.0f / 127.0f) - 1.0f;
    const float F0 = gx * gx;   // pairs with a
    const float F1 = gy * gy;   // pairs with b
    const float F2 = gx * gy;   // pairs with 2c
    const float F3 = gx;
    const float F4 = gy;
    const float F5 = 1.0f;      // F6 = F7 = 0 (K padding)

    // 16x4 f32 A layout: lanes 0-15 -> K={0,1}, lanes 16-31 -> K={2,3}
    v2f va, va2;
    va.x  = hi ? F2 : F0;       // K=0 / K=2
    va.y  = hi ? F3 : F1;       // K=1 / K=3
    va2.x = hi ? 0.0f : F4;     // K=4 / K=6(pad)
    va2.y = hi ? 0.0f : F5;     // K=5 / K=7(pad)

    const float S = -0.72134752044448170f;  // -0.5 * log2(e): folded so WMMA feeds exp2 directly

    float acc[8];
#pragma unroll
    for (int k = 0; k < 8; ++k) acc[k] = 0.0f;

    const unsigned gbase = img * (unsigned)N_GAUSS;

    for (unsigned gt = 0; gt < 8; ++gt) {
        const unsigned n  = gbase + gt * 16u + m;   // both half-waves load same 16 gaussians
        const float cx  = centers[n * 2 + 0];
        const float cy  = centers[n * 2 + 1];
        const float a   = covs[n * 3 + 0];
        const float b   = covs[n * 3 + 1];
        const float cc  = covs[n * 3 + 2];
        const float amp = amps[n];

        // Rank-6 expansion of a*dx^2 + b*dy^2 + 2c*dx*dy, scaled by -0.5*log2(e)
        const float G0 = S * a;
        const float G1 = S * b;
        const float G2 = S * (2.0f * cc);
        const float G3 = S * (-2.0f * (a * cx + cc * cy));
        const float G4 = S * (-2.0f * (b * cy + cc * cx));
        const float G5 = S * (a * cx * cx + b * cy * cy + 2.0f * cc * cx * cy);

        // 4x16 f32 B layout: lanes 0-15 -> K={0,1}, lanes 16-31 -> K={2,3}
        v2f vb, vb2;
        vb.x  = hi ? G2 : G0;
        vb.y  = hi ? G3 : G1;
        vb2.x = hi ? 0.0f : G4;
        vb2.y = hi ? 0.0f : G5;

        // Q = F * G^T  (f32 accumulate, exact)
        v8f q = {};
        q = __builtin_amdgcn_wmma_f32_16x16x4_f32(false, va,  false, vb,  (short)0, q, false, false);
        q = __builtin_amdgcn_wmma_f32_16x16x4_f32(false, va2, false, vb2, (short)0, q, false, false);

        // exp2 (v_exp_f32) and accumulate amp-weighted contribution per (M,N) element
#pragma unroll
        for (int k = 0; k < 8; ++k)
            acc[k] += __builtin_amdgcn_exp2f(q[k]) * amp;
    }

    // Reduce over the N dimension: 16-lane butterfly (stays inside each half-wave group)
#pragma unroll
    for (int k = 0; k < 8; ++k) {
        float v = acc[k];
        v += __shfl_xor(v, 1, 32);
        v += __shfl_xor(v, 2, 32);
        v += __shfl_xor(v, 4, 32);
        v += __shfl_xor(v, 8, 32);
        acc[k] = v;
    }

    // D layout: acc[k] group0 -> pixel (c0+k), group1 -> pixel (c0+8+k).
    // Stage through LDS so the final store is one coalesced 16-lane b32 store.
    float* st = &stage[waveInBlk * 16u];
#pragma unroll
    for (int k = 0; k < 8; ++k) {
        if (lane == (unsigned)k)        st[k]     = acc[k];
        if (lane == (unsigned)(16 + k)) st[8 + k] = acc[k];
    }
    __syncthreads();
    if (lane < 16u) {
        out[img * (ROWS * COLS) + row * COLS + c0 + lane] = st[lane];
    }
}

extern "C" void kernel_launch(void* const* d_in, const int* in_sizes, int n_in,
                              void* d_out, int out_size, void* d_ws, size_t ws_size,
                              hipStream_t stream) {
    (void)in_sizes; (void)n_in; (void)out_size; (void)d_ws; (void)ws_size;
    const float* centers = (const float*)d_in[0];   // (32,128,2) f32
    const float* covs    = (const float*)d_in[1];   // (32,128,3) f32
    const float* amps    = (const float*)d_in[2];   // (32,128,1) f32
    float* out           = (float*)d_out;           // (32,128,128) f32

    // 32768 waves total: 32 images x 1024 pixel tiles; 8 waves per 256-thread block.
    rbf_wmma_kernel<<<4096, 256, 0, stream>>>(centers, covs, amps, out);
}